// AttentionBlock_300647711321
// MI455X (gfx1250) — compile-verified
//
#include <hip/hip_runtime.h>
#include <hip/hip_bf16.h>

typedef float f32x2 __attribute__((ext_vector_type(2)));
typedef float f32x8 __attribute__((ext_vector_type(8)));
typedef int   i32x4v __attribute__((vector_size(16)));   // matches builtin param

#define AS1 __attribute__((address_space(1)))
#define AS3 __attribute__((address_space(3)))

#if defined(__has_builtin)
#if __has_builtin(__builtin_amdgcn_global_load_async_to_lds_b128)
#define HAVE_ASYNC_LDS 1
#endif
#endif
#ifndef HAVE_ASYNC_LDS
#define HAVE_ASYNC_LDS 0
#endif

// Async global->LDS 16-byte copy (gfx1250 GLOBAL_LOAD_ASYNC_TO_LDS_B128,
// tracked by ASYNCcnt). Falls back to a synchronous copy if the builtin is
// absent; the surrounding double-buffer structure stays correct either way.
__device__ __forceinline__ void cp_async_b128(void* lds, const void* g) {
#if HAVE_ASYNC_LDS
    __builtin_amdgcn_global_load_async_to_lds_b128(
        (AS1 i32x4v*)g, (AS3 i32x4v*)lds, 0, 0);
#else
    *(float4*)lds = *(const float4*)g;
#endif
}

__device__ __forceinline__ void cp_async_wait() {
#if HAVE_ASYNC_LDS
#if __has_builtin(__builtin_amdgcn_s_wait_asynccnt)
    __builtin_amdgcn_s_wait_asynccnt(0);
#else
    asm volatile("s_wait_asynccnt 0x0" ::: "memory");
#endif
#endif
}

__device__ __forceinline__ f32x8 wmma_f32(f32x2 a, f32x2 b, f32x8 c) {
    // D = A(16x4) * B(4x16) + C(16x16), fp32
    return __builtin_amdgcn_wmma_f32_16x16x4_f32(false, a, false, b, (short)0, c, false, false);
}

// ---------------------------------------------------------------------------
// Kernel 1: GroupNorm.  One block per (batch, group). C=512, GROUPS=32 -> 16
// channels * 1024 spatial = 16384 floats per group.
// ---------------------------------------------------------------------------
__global__ __launch_bounds__(256) void groupnorm_kernel(
    const float* __restrict__ x, const float* __restrict__ gw,
    const float* __restrict__ gb, float* __restrict__ xn)
{
    const int g     = blockIdx.x;        // 0..255
    const int batch = g >> 5;
    const int grp   = g & 31;
    const float* xp = x  + ((size_t)batch * 512 + grp * 16) * 1024;
    float*       op = xn + ((size_t)batch * 512 + grp * 16) * 1024;
    const int tid = threadIdx.x;

    float s = 0.f, ss = 0.f;
    for (int i = tid; i < 4096; i += 256) {          // 4096 float4 = 16384 floats
        float4 v = ((const float4*)xp)[i];
        s  += v.x + v.y + v.z + v.w;
        ss += v.x * v.x + v.y * v.y + v.z * v.z + v.w * v.w;
    }
    __shared__ float rs[256], rss[256];
    rs[tid] = s; rss[tid] = ss;
    __syncthreads();
    for (int st = 128; st > 0; st >>= 1) {
        if (tid < st) { rs[tid] += rs[tid + st]; rss[tid] += rss[tid + st]; }
        __syncthreads();
    }
    const float mean = rs[0] * (1.f / 16384.f);
    const float var  = rss[0] * (1.f / 16384.f) - mean * mean;
    const float inv  = rsqrtf(var + 1e-5f);

    for (int i = tid; i < 4096; i += 256) {
        float4 v = ((const float4*)xp)[i];
        const int c = grp * 16 + (i >> 8);           // channel of this float4
        const float sc = gw[c] * inv;
        const float sb = gb[c] - mean * sc;
        float4 o;
        o.x = v.x * sc + sb; o.y = v.y * sc + sb;
        o.z = v.z * sc + sb; o.w = v.w * sc + sb;
        ((float4*)op)[i] = o;
    }
}

// ---------------------------------------------------------------------------
// Kernel 2: batched GEMM  C[b] = A * X[b] + bias (+ residual[b])
// A: [M,K] row-major (weights), X: [b,K,N], C: [b,M,N].
// Block tile 64x128, K slab 16, double-buffered async LDS staging.
// Wave w: m-tile (w&3)*16, n-range (w>>2)*64 (4 tiles of 16).
// ---------------------------------------------------------------------------
#define BM 64
#define BN 128
#define KB 16
#define APAD 20   // 80B row stride: 16B aligned, conflict-free column reads

__global__ __launch_bounds__(256) void gemm_bias_kernel(
    const float* __restrict__ A, const float* __restrict__ Bm,
    const float* __restrict__ bias, const float* __restrict__ res,
    float* __restrict__ C, int M, int N, int K)
{
    __shared__ float As[2][BM][APAD];
    __shared__ float Bs[2][KB][BN];

    const int bz = blockIdx.z;
    const float* Bp = Bm + (size_t)bz * K * N;
    float*       Cp = C  + (size_t)bz * M * N;
    const float* Rp = res ? res + (size_t)bz * M * N : nullptr;

    const int m0 = blockIdx.y * BM;
    const int n0 = blockIdx.x * BN;
    const int tid = threadIdx.x;
    const int w = tid >> 5, lane = tid & 31;
    const int l15 = lane & 15, hi = lane >> 4;
    const int mw = (w & 3) * 16;         // wave m offset in block tile
    const int nw = (w >> 2) * 64;        // wave n offset (4 tiles of 16)

    auto stage = [&](int k0, int buf) {
        // A tile: 64x16 floats, one b128 per thread
        const int r  = tid >> 2;
        const int c4 = (tid & 3) << 2;
        cp_async_b128(&As[buf][r][c4], A + (size_t)(m0 + r) * K + k0 + c4);
        // B tile: 16x128 floats, two b128 per thread
        #pragma unroll
        for (int i = 0; i < 2; ++i) {
            const int idx = tid + i * 256;
            const int rr  = idx >> 5;
            const int cc4 = (idx & 31) << 2;
            cp_async_b128(&Bs[buf][rr][cc4], Bp + (size_t)(k0 + rr) * N + n0 + cc4);
        }
    };

    f32x8 acc[4] = {};

    stage(0, 0);
    cp_async_wait();
    __syncthreads();

    int buf = 0;
    for (int k0 = 0; k0 < K; k0 += KB) {
        if (k0 + KB < K) stage(k0 + KB, buf ^ 1);   // prefetch next slab (async)

        #pragma unroll
        for (int kk = 0; kk < KB; kk += 4) {
            f32x2 a;
            a.x = As[buf][mw + l15][kk + 2 * hi];
            a.y = As[buf][mw + l15][kk + 1 + 2 * hi];
            #pragma unroll
            for (int j = 0; j < 4; ++j) {
                f32x2 b;
                b.x = Bs[buf][kk + 2 * hi][nw + j * 16 + l15];
                b.y = Bs[buf][kk + 1 + 2 * hi][nw + j * 16 + l15];
                acc[j] = wmma_f32(a, b, acc[j]);
            }
        }
        cp_async_wait();
        __syncthreads();
        buf ^= 1;
    }

    #pragma unroll
    for (int r = 0; r < 8; ++r) {
        const int row = m0 + mw + r + 8 * hi;
        const float bv = bias[row];
        #pragma unroll
        for (int j = 0; j < 4; ++j) {
            const int col = n0 + nw + j * 16 + l15;
            const size_t off = (size_t)row * N + col;
            float v = acc[j][r] + bv;
            if (Rp) v += Rp[off];
            Cp[off] = v;
        }
    }
}

// ---------------------------------------------------------------------------
// Kernel 3: flash attention. grid = (T/64, B*HEADS). Head dim 64, T=1024.
// qkv layout per (b,head): q at rows head*192+0..63, k +64, v +128 (each [64,T]).
// K/V s-tiles double-buffered via async LDS copies: prefetch of tile i+1
// overlaps S-GEMM + softmax + PV-GEMM of tile i.
// ---------------------------------------------------------------------------
#define KVPAD 36  // 144B row stride: 16B aligned, conflict-free row reads

__global__ __launch_bounds__(256) void attn_kernel(
    const float* __restrict__ qkv, float* __restrict__ hout)
{
    __shared__ float Qs[64][65];        // Q^T * (1/8), [t][c]
    __shared__ float Ks[2][64][KVPAD];  // [c][s]
    __shared__ float Vs[2][64][KVPAD];  // [c][s]
    __shared__ float Ss[64][33];        // scores / probs [t][s]
    __shared__ float mrow[64], lrow[64], frow[64];

    const int bh = blockIdx.y;
    const int batch = bh >> 3, head = bh & 7;
    const float* base = qkv + ((size_t)batch * 1536 + head * 192) * 1024;
    const float* qp = base;
    const float* kp = base + 64 * 1024;
    const float* vp = base + 128 * 1024;
    const int t0  = blockIdx.x * 64;
    const int tid = threadIdx.x;
    const int w = tid >> 5, lane = tid & 31;
    const int l15 = lane & 15, hi = lane >> 4;
    const int mw = (w & 3) * 16;       // wave t-tile
    const int nS = (w >> 2) * 16;      // wave s-tile within 32-wide slab
    const int nO = (w >> 2) * 32;      // wave c-range for O (2 tiles of 16)
    const float sc2 = 0.125f;          // (1/sqrt(sqrt(64)))^2 folded into Q

    auto stage_kv = [&](int s0, int buf) {   // 64x32 floats each, async b128
        #pragma unroll
        for (int i = 0; i < 2; ++i) {
            const int idx = tid + i * 256;
            const int c  = idx >> 3;
            const int s4 = (idx & 7) << 2;
            cp_async_b128(&Ks[buf][c][s4], kp + (size_t)c * 1024 + s0 + s4);
            cp_async_b128(&Vs[buf][c][s4], vp + (size_t)c * 1024 + s0 + s4);
        }
    };

    stage_kv(0, 0);
    // load Q tile transposed + scaled (needs transpose+scale -> regular path)
    for (int i = tid; i < 1024; i += 256) {     // 1024 float4 loads
        const int c  = i >> 4;
        const int t4 = (i & 15) << 2;
        float4 v = *(const float4*)(qp + (size_t)c * 1024 + t0 + t4);
        Qs[t4 + 0][c] = v.x * sc2; Qs[t4 + 1][c] = v.y * sc2;
        Qs[t4 + 2][c] = v.z * sc2; Qs[t4 + 3][c] = v.w * sc2;
    }
    if (tid < 64) { mrow[tid] = -3.0e38f; lrow[tid] = 0.f; }
    cp_async_wait();
    __syncthreads();

    f32x8 oacc[2] = {};
    int buf = 0;

    for (int s0 = 0; s0 < 1024; s0 += 32) {
        if (s0 + 32 < 1024) stage_kv(s0 + 32, buf ^ 1);  // async prefetch

        // S = Q^T K : one 16x16 tile per wave, K-dim (channels) = 64
        f32x8 sacc = {};
        #pragma unroll
        for (int kk = 0; kk < 64; kk += 4) {
            f32x2 a, b;
            a.x = Qs[mw + l15][kk + 2 * hi];
            a.y = Qs[mw + l15][kk + 1 + 2 * hi];
            b.x = Ks[buf][kk + 2 * hi][nS + l15];
            b.y = Ks[buf][kk + 1 + 2 * hi][nS + l15];
            sacc = wmma_f32(a, b, sacc);
        }
        #pragma unroll
        for (int r = 0; r < 8; ++r)
            Ss[mw + r + 8 * hi][nS + l15] = sacc[r];
        __syncthreads();

        // online softmax row update (one thread per t-row)
        if (tid < 64) {
            const float m_old = mrow[tid];
            float mx = m_old;
            for (int j = 0; j < 32; ++j) mx = fmaxf(mx, Ss[tid][j]);
            const float f = __expf(m_old - mx);
            float sum = 0.f;
            for (int j = 0; j < 32; ++j) {
                const float p = __expf(Ss[tid][j] - mx);
                Ss[tid][j] = p; sum += p;
            }
            mrow[tid] = mx;
            lrow[tid] = lrow[tid] * f + sum;
            frow[tid] = f;
        }
        __syncthreads();

        // rescale O accumulators by row factor (C-layout row = mw + r + 8*hi)
        #pragma unroll
        for (int r = 0; r < 8; ++r) {
            const float f = frow[mw + r + 8 * hi];
            oacc[0][r] *= f; oacc[1][r] *= f;
        }
        // O += P * V^T : A = Ss[t][s] (K = s = 32), B[k=s][n=c] = Vs[c][s]
        #pragma unroll
        for (int kk = 0; kk < 32; kk += 4) {
            f32x2 a;
            a.x = Ss[mw + l15][kk + 2 * hi];
            a.y = Ss[mw + l15][kk + 1 + 2 * hi];
            #pragma unroll
            for (int j = 0; j < 2; ++j) {
                const int cb = nO + j * 16;
                f32x2 b;
                b.x = Vs[buf][cb + l15][kk + 2 * hi];
                b.y = Vs[buf][cb + l15][kk + 1 + 2 * hi];
                oacc[j] = wmma_f32(a, b, oacc[j]);
            }
        }
        cp_async_wait();
        __syncthreads();
        buf ^= 1;
    }

    // finalize: O / l  -> stage into Qs (free now) for coalesced store
    #pragma unroll
    for (int j = 0; j < 2; ++j) {
        #pragma unroll
        for (int r = 0; r < 8; ++r) {
            const int trow = mw + r + 8 * hi;
            Qs[trow][nO + j * 16 + l15] = oacc[j][r] / lrow[trow];
        }
    }
    __syncthreads();
    // h[batch, head*64 + c, t0 + t]
    float* op = hout + ((size_t)batch * 512 + head * 64) * 1024 + t0;
    for (int i = tid; i < 4096; i += 256) {
        const int c = i >> 6, t = i & 63;
        op[(size_t)c * 1024 + t] = Qs[t][c];
    }
}

// ---------------------------------------------------------------------------
extern "C" void kernel_launch(void* const* d_in, const int* in_sizes, int n_in,
                              void* d_out, int out_size, void* d_ws, size_t ws_size,
                              hipStream_t stream)
{
    const float* x      = (const float*)d_in[0];
    const float* gn_w   = (const float*)d_in[1];
    const float* gn_b   = (const float*)d_in[2];
    const float* qkv_w  = (const float*)d_in[3];
    const float* qkv_b  = (const float*)d_in[4];
    const float* proj_w = (const float*)d_in[5];
    const float* proj_b = (const float*)d_in[6];
    float* out = (float*)d_out;
    float* ws  = (float*)d_ws;

    float* xn   = ws;                           // 8*512*1024 floats (16 MB), reused for attn out
    float* qkvb = ws + (size_t)8 * 512 * 1024;  // 8*1536*1024 floats (48 MB)

    // 1. GroupNorm
    groupnorm_kernel<<<dim3(256), dim3(256), 0, stream>>>(x, gn_w, gn_b, xn);

    // 2. QKV GEMM: [1536,512] x [b,512,1024]
    gemm_bias_kernel<<<dim3(1024 / BN, 1536 / BM, 8), dim3(256), 0, stream>>>(
        qkv_w, xn, qkv_b, nullptr, qkvb, 1536, 1024, 512);

    // 3. Attention (writes h into xn region, which is free now)
    attn_kernel<<<dim3(16, 64), dim3(256), 0, stream>>>(qkvb, xn);

    // 4. Proj GEMM + bias + residual
    gemm_bias_kernel<<<dim3(1024 / BN, 512 / BM, 8), dim3(256), 0, stream>>>(
        proj_w, xn, proj_b, x, out, 512, 1024, 512);
}